// Attention_3917010174340
// MI455X (gfx1250) — compile-verified
//
#include <hip/hip_runtime.h>
#include <hip/hip_bf16.h>
#include <math.h>

// Problem constants (from reference setup_inputs)
#define B_   32
#define T_   2048
#define H_   1024
#define OUT_ 256

typedef __attribute__((ext_vector_type(2))) float v2f;
typedef __attribute__((ext_vector_type(8))) float v8f;

// D(16x16,f32) = A(16x4,f32) x B(4x16,f32) + C  — CDNA5 V_WMMA_F32_16X16X4_F32
__device__ __forceinline__ v8f wmma4(v2f a, v2f b, v8f c) {
  return __builtin_amdgcn_wmma_f32_16x16x4_f32(
      /*neg_a=*/false, a, /*neg_b=*/false, b,
      /*c_mod=*/(short)0, c, /*reuse_a=*/false, /*reuse_b=*/false);
}

// ---------------------------------------------------------------------------
// Kernel 1: h_t[b,:] = hidden[b, T-1, :] @ W1 + b1      (32x1024 @ 1024x1024)
// One wave32 per 16x16 tile. A(16x4): lanes 0..15 M=lane K=0,1; lanes 16..31
// K=2,3. B(4x16): lanes 0..15 N=lane K=0,1; lanes 16..31 K=2,3.
// C/D: vgpr r, lane l -> M = r + 8*(l>>4), N = l&15.
// ---------------------------------------------------------------------------
__global__ __launch_bounds__(32) void k_ht(const float* __restrict__ hid,
                                           const float* __restrict__ W1,
                                           const float* __restrict__ b1,
                                           float* __restrict__ ht) {
  const int lane = threadIdx.x;
  const int n0 = blockIdx.x * 16;
  const int m0 = blockIdx.y * 16;
  const int lm = lane & 15;
  const int kh = (lane >> 4) << 1;   // 0 or 2: this lane's K-pair base
  const float* arow = hid + ((size_t)(m0 + lm) * T_ + (T_ - 1)) * H_;
  v8f acc = {};
  for (int kb = 0; kb < H_; kb += 4) {
    v2f a, b;
    a.x = arow[kb + kh];
    a.y = arow[kb + kh + 1];
    b.x = W1[(size_t)(kb + kh)     * H_ + n0 + lm];
    b.y = W1[(size_t)(kb + kh + 1) * H_ + n0 + lm];
    acc = wmma4(a, b, acc);
  }
  const int cn   = n0 + lm;
  const int mb   = m0 + ((lane >> 4) << 3);
  const float bi = b1[cn];
#pragma unroll
  for (int r = 0; r < 8; ++r)
    ht[(size_t)(mb + r) * H_ + cn] = acc[r] + bi;
}

// ---------------------------------------------------------------------------
// Kernel 2: u[b,j] = sum_k W1[j,k] * h_t[b,k]  == (h_t @ W1^T)   32x1024
// (The b1.h_t score offset is batch-constant and softmax is shift-invariant,
//  so it is dropped entirely.)
// ---------------------------------------------------------------------------
__global__ __launch_bounds__(32) void k_u(const float* __restrict__ ht,
                                          const float* __restrict__ W1,
                                          float* __restrict__ u) {
  const int lane = threadIdx.x;
  const int n0 = blockIdx.x * 16;
  const int m0 = blockIdx.y * 16;
  const int lm = lane & 15;
  const int kh = (lane >> 4) << 1;
  const float* arow = ht + (size_t)(m0 + lm) * H_;
  const float* brow = W1 + (size_t)(n0 + lm) * H_;   // B[k][n] = W1[n][k]
  v8f acc = {};
  for (int kb = 0; kb < H_; kb += 4) {
    v2f a, b;
    a.x = arow[kb + kh];
    a.y = arow[kb + kh + 1];
    b.x = brow[kb + kh];
    b.y = brow[kb + kh + 1];
    acc = wmma4(a, b, acc);
  }
  const int cn = n0 + lm;
  const int mb = m0 + ((lane >> 4) << 3);
#pragma unroll
  for (int r = 0; r < 8; ++r)
    u[(size_t)(mb + r) * H_ + cn] = acc[r];
}

// ---------------------------------------------------------------------------
// Kernel 3 (FUSED, single pass over hidden_states = 256 MB):
// flash-style online softmax + weighted context accumulation.
// One wave32 per chunk of T_/SEGS rows of one batch. Zero __syncthreads,
// zero LDS: the lane's 32-float row slice is used for both the score dot
// (closed with a shfl_xor butterfly) and the context accumulation, which
// lives in 8 float4 registers per lane. Running-max rescale is skipped
// behind a wave-uniform branch when the max doesn't change.
// Outputs per wave: c_part[1024] (unnormalized, local max), m, s.
// ---------------------------------------------------------------------------
template <int SEGS>
__global__ __launch_bounds__(256) void k_fused(const float* __restrict__ hid,
                                               const float* __restrict__ u,
                                               float* __restrict__ part_c,
                                               float* __restrict__ part_ms) {
  const int lane = threadIdx.x & 31;
  const int wave = threadIdx.x >> 5;
  const int seg  = blockIdx.x * 8 + wave;        // 0 .. B_*SEGS-1
  const int b    = seg / SEGS;
  const int t0   = (seg % SEGS) * (T_ / SEGS);

  const float4* up = (const float4*)(u + (size_t)b * H_);
  float4 uv[8];
#pragma unroll
  for (int i = 0; i < 8; ++i) uv[i] = up[lane + 32 * i];

  float4 cacc[8];
#pragma unroll
  for (int i = 0; i < 8; ++i) cacc[i] = make_float4(0.f, 0.f, 0.f, 0.f);
  float m = -1e30f, s = 0.f;

  const float4* hp0 = (const float4*)(hid + ((size_t)b * T_ + t0) * H_);
  for (int r = 0; r < T_ / SEGS; ++r) {
    const float4* hp = hp0 + (size_t)r * (H_ / 4);
    float4 hv[8];
#pragma unroll
    for (int i = 0; i < 8; ++i) hv[i] = hp[lane + 32 * i];

    float d = 0.f;
#pragma unroll
    for (int i = 0; i < 8; ++i)
      d += hv[i].x * uv[i].x + hv[i].y * uv[i].y +
           hv[i].z * uv[i].z + hv[i].w * uv[i].w;
#pragma unroll
    for (int off = 16; off; off >>= 1) d += __shfl_xor(d, off, 32);

    const float mn    = fmaxf(m, d);
    const float alpha = __expf(m - mn);          // 0 on first row, 1 steady-state
    const float w     = __expf(d - mn);
    s = s * alpha + w;
    if (alpha != 1.0f) {                          // wave-uniform branch
#pragma unroll
      for (int i = 0; i < 8; ++i) {
        cacc[i].x *= alpha; cacc[i].y *= alpha;
        cacc[i].z *= alpha; cacc[i].w *= alpha;
      }
    }
#pragma unroll
    for (int i = 0; i < 8; ++i) {
      cacc[i].x = fmaf(w, hv[i].x, cacc[i].x);
      cacc[i].y = fmaf(w, hv[i].y, cacc[i].y);
      cacc[i].z = fmaf(w, hv[i].z, cacc[i].z);
      cacc[i].w = fmaf(w, hv[i].w, cacc[i].w);
    }
    m = mn;
  }

  float4* cp = (float4*)(part_c + (size_t)seg * H_);
#pragma unroll
  for (int i = 0; i < 8; ++i) cp[lane + 32 * i] = cacc[i];
  if (lane == 0) {
    part_ms[2 * seg]     = m;
    part_ms[2 * seg + 1] = s;
  }
}

// ---------------------------------------------------------------------------
// Kernel 4: combine per-wave partials into the context half of pre_activation:
// pre[b][h] = sum_g exp(m_g - M) * c_g[h] / sum_g exp(m_g - M) * s_g
// Deterministic, no atomics. grid = (H/256, B).
// ---------------------------------------------------------------------------
template <int SEGS>
__global__ __launch_bounds__(256) void k_combine(const float* __restrict__ part_c,
                                                 const float* __restrict__ part_ms,
                                                 float* __restrict__ pre) {
  const int h = blockIdx.x * 256 + threadIdx.x;
  const int b = blockIdx.y;
  float M = -1e30f;
#pragma unroll 4
  for (int g = 0; g < SEGS; ++g)
    M = fmaxf(M, part_ms[2 * (b * SEGS + g)]);
  float S = 0.f, c = 0.f;
  for (int g = 0; g < SEGS; ++g) {
    const int idx = b * SEGS + g;
    const float e = __expf(part_ms[2 * idx] - M);
    S += e * part_ms[2 * idx + 1];
    c  = fmaf(e, part_c[(size_t)idx * H_ + h], c);
  }
  pre[(size_t)b * (2 * H_) + h] = c / S;
}

// ---------------------------------------------------------------------------
// Kernel 5: copy h_t into the second half of pre_activation.
// ---------------------------------------------------------------------------
__global__ __launch_bounds__(256) void k_copyht(const float* __restrict__ ht,
                                                float* __restrict__ pre) {
  const int i = blockIdx.x * 256 + threadIdx.x;   // 0 .. B_*H_-1
  const int b = i >> 10;
  const int k = i & (H_ - 1);
  pre[(size_t)b * (2 * H_) + H_ + k] = ht[i];
}

// ---------------------------------------------------------------------------
// Kernel 6: out = tanh(pre(32x2048) @ Wv(2048x256) + bv)   — WMMA f32
// ---------------------------------------------------------------------------
__global__ __launch_bounds__(32) void k_out(const float* __restrict__ pre,
                                            const float* __restrict__ Wv,
                                            const float* __restrict__ bv,
                                            float* __restrict__ out) {
  const int lane = threadIdx.x;
  const int n0 = blockIdx.x * 16;
  const int m0 = blockIdx.y * 16;
  const int lm = lane & 15;
  const int kh = (lane >> 4) << 1;
  const float* arow = pre + (size_t)(m0 + lm) * (2 * H_);
  v8f acc = {};
  for (int kb = 0; kb < 2 * H_; kb += 4) {
    v2f a, b;
    a.x = arow[kb + kh];
    a.y = arow[kb + kh + 1];
    b.x = Wv[(size_t)(kb + kh)     * OUT_ + n0 + lm];
    b.y = Wv[(size_t)(kb + kh + 1) * OUT_ + n0 + lm];
    acc = wmma4(a, b, acc);
  }
  const int cn   = n0 + lm;
  const int mb   = m0 + ((lane >> 4) << 3);
  const float bi = bv[cn];
#pragma unroll
  for (int r = 0; r < 8; ++r)
    out[(size_t)(mb + r) * OUT_ + cn] = tanhf(acc[r] + bi);
}

// ---------------------------------------------------------------------------
extern "C" void kernel_launch(void* const* d_in, const int* in_sizes, int n_in,
                              void* d_out, int out_size, void* d_ws, size_t ws_size,
                              hipStream_t stream) {
  const float* hid = (const float*)d_in[0];   // (32, 2048, 1024)
  const float* W1  = (const float*)d_in[1];   // (1024, 1024)
  const float* b1  = (const float*)d_in[2];   // (1024,)
  const float* Wv  = (const float*)d_in[3];   // (2048, 256)
  const float* bv  = (const float*)d_in[4];   // (256,)
  float* out = (float*)d_out;                 // (32, 256)

  // Workspace layout (floats): ht | u | pre | part_ms | part_c
  float* ws  = (float*)d_ws;
  float* ht  = ws;                            // B*H
  float* u   = ht + (size_t)B_ * H_;          // B*H
  float* pre = u + (size_t)B_ * H_;           // B*2H
  float* ms  = pre + (size_t)B_ * 2 * H_;     // B*SEGS*2
  const size_t fixedf = (size_t)B_ * H_ * 2 + (size_t)B_ * 2 * H_;

  k_ht<<<dim3(H_ / 16, B_ / 16), 32, 0, stream>>>(hid, W1, b1, ht);
  k_u <<<dim3(H_ / 16, B_ / 16), 32, 0, stream>>>(ht, W1, u);

  // Pick the largest partial buffer that fits in d_ws (deterministic per ws_size).
  const size_t need64 = (fixedf + (size_t)B_ * 64 * 2 + (size_t)B_ * 64 * H_) * sizeof(float);
  const size_t need16 = (fixedf + (size_t)B_ * 16 * 2 + (size_t)B_ * 16 * H_) * sizeof(float);
  if (ws_size >= need64) {
    constexpr int SEGS = 64;                  // 2048 waves, 32 rows/wave, 8 MB partials
    float* pc = ms + (size_t)B_ * SEGS * 2;
    k_fused<SEGS>  <<<B_ * SEGS / 8, 256, 0, stream>>>(hid, u, pc, ms);
    k_combine<SEGS><<<dim3(H_ / 256, B_), 256, 0, stream>>>(pc, ms, pre);
  } else if (ws_size >= need16) {
    constexpr int SEGS = 16;                  // 512 waves, 128 rows/wave, 2 MB partials
    float* pc = ms + (size_t)B_ * SEGS * 2;
    k_fused<SEGS>  <<<B_ * SEGS / 8, 256, 0, stream>>>(hid, u, pc, ms);
    k_combine<SEGS><<<dim3(H_ / 256, B_), 256, 0, stream>>>(pc, ms, pre);
  } else {
    constexpr int SEGS = 4;                   // minimal footprint fallback
    float* pc = ms + (size_t)B_ * SEGS * 2;
    k_fused<SEGS>  <<<B_ * SEGS / 8, 256, 0, stream>>>(hid, u, pc, ms);
    k_combine<SEGS><<<dim3(H_ / 256, B_), 256, 0, stream>>>(pc, ms, pre);
  }

  k_copyht<<<B_ * H_ / 256, 256, 0, stream>>>(ht, pre);
  k_out<<<dim3(OUT_ / 16, B_ / 16), 32, 0, stream>>>(pre, Wv, bv, out);
}